// AttnHGCN_67456756351000
// MI455X (gfx1250) — compile-verified
//
#include <hip/hip_runtime.h>
#include <hip/hip_bf16.h>
#include <math.h>

typedef float v2f __attribute__((ext_vector_type(2)));
typedef float v8f __attribute__((ext_vector_type(8)));

#define N_USERS     100000
#define N_ENTITIES  200000
#define CHANNEL     64
#define N_REL_W     9          // N_RELATIONS - 1
#define N_FACTORS   4
#define N_KG        2000000
#define N_INTER     1000000

// ---------------- utility kernels ----------------

__global__ void k_zero_f32(float* __restrict__ p, long long n) {
    long long i = (long long)blockIdx.x * blockDim.x + threadIdx.x;
    long long stride = (long long)gridDim.x * blockDim.x;
    for (; i < n; i += stride) p[i] = 0.0f;
}

__global__ void k_copy_f32(float* __restrict__ dst, const float* __restrict__ src, long long n) {
    long long i = (long long)blockIdx.x * blockDim.x + threadIdx.x;
    long long stride = (long long)gridDim.x * blockDim.x;
    for (; i < n; i += stride) dst[i] = src[i];
}

// counts[h] += 1 for each KG edge head h  (scatter_mean denominator)
__global__ void k_count_heads(const int* __restrict__ head, float* __restrict__ counts) {
    long long i = (long long)blockIdx.x * blockDim.x + threadIdx.x;
    if (i < N_KG) atomicAdd(&counts[head[i]], 1.0f);
}

// disen_w = softmax(att, axis=-1) @ weight ;  cor = sum_{i<j} cos^2(att_i, att_j)
__global__ void k_disen_and_cor(const float* __restrict__ att,     // [4][9]
                                const float* __restrict__ weight,  // [9][64]
                                float* __restrict__ disen_w,       // [4][64]
                                float* __restrict__ cor_out) {
    int c = threadIdx.x;  // 64 threads
    for (int f = 0; f < N_FACTORS; ++f) {
        float m = -1e30f;
        for (int r = 0; r < N_REL_W; ++r) m = fmaxf(m, att[f * N_REL_W + r]);
        float e[N_REL_W]; float s = 0.f;
        for (int r = 0; r < N_REL_W; ++r) { e[r] = __expf(att[f * N_REL_W + r] - m); s += e[r]; }
        float acc = 0.f;
        for (int r = 0; r < N_REL_W; ++r) acc += (e[r] / s) * weight[r * CHANNEL + c];
        disen_w[f * CHANNEL + c] = acc;
    }
    if (c == 0) {
        float nrm[N_FACTORS];
        for (int f = 0; f < N_FACTORS; ++f) {
            float s = 0.f;
            for (int r = 0; r < N_REL_W; ++r) s += att[f * N_REL_W + r] * att[f * N_REL_W + r];
            nrm[f] = sqrtf(s);
        }
        float cor = 0.f;
        for (int i = 0; i < N_FACTORS; ++i)
            for (int j = i + 1; j < N_FACTORS; ++j) {
                float d = 0.f;
                for (int r = 0; r < N_REL_W; ++r) d += att[i * N_REL_W + r] * att[j * N_REL_W + r];
                float g = d / (nrm[i] * nrm[j]);
                cor += g * g;
            }
        *cor_out = cor;
    }
}

// ---------------- scatter kernels (L2-resident atomics) ----------------

// eagg[head] += entity_src[tail] * weight[etype-1]; one wave32 per edge, 2 ch/lane
__global__ void k_kg_scatter(const int* __restrict__ head, const int* __restrict__ tail,
                             const int* __restrict__ etype,
                             const float* __restrict__ esrc,   // [N_ENTITIES][64]
                             const float* __restrict__ weight, // [9][64]
                             float* __restrict__ eagg) {
    long long t = (long long)blockIdx.x * blockDim.x + threadIdx.x;
    long long e = t >> 5;
    if (e >= N_KG) return;
    int lane = (int)(t & 31);
    int h = head[e], tl = tail[e], r = etype[e] - 1;
    int c = lane * 2;
    float2 v = *(const float2*)(esrc + (long long)tl * CHANNEL + c);
    float2 w = *(const float2*)(weight + (long long)r * CHANNEL + c);
    atomicAdd(&eagg[(long long)h * CHANNEL + c + 0], v.x * w.x);
    atomicAdd(&eagg[(long long)h * CHANNEL + c + 1], v.y * w.y);
}

// uagg[u] += inter_w * entity_src[i]
__global__ void k_inter_scatter(const int* __restrict__ uix, const int* __restrict__ iix,
                                const float* __restrict__ ew,
                                const float* __restrict__ esrc,
                                float* __restrict__ uagg) {
    long long t = (long long)blockIdx.x * blockDim.x + threadIdx.x;
    long long e = t >> 5;
    if (e >= N_INTER) return;
    int lane = (int)(t & 31);
    int u = uix[e], it = iix[e];
    float w = ew[e];
    int c = lane * 2;
    float2 v = *(const float2*)(esrc + (long long)it * CHANNEL + c);
    atomicAdd(&uagg[(long long)u * CHANNEL + c + 0], w * v.x);
    atomicAdd(&uagg[(long long)u * CHANNEL + c + 1], w * v.y);
}

// ---------------- finalize kernels ----------------

// entity: agg /= max(count,1); l2-normalize row; eagg <- normalized (new emb); eres += normalized
__global__ void k_entity_finalize(float* __restrict__ eagg, const float* __restrict__ counts,
                                  float* __restrict__ eres) {
    int wave = blockIdx.x * (blockDim.x >> 5) + (threadIdx.x >> 5);
    if (wave >= N_ENTITIES) return;
    int lane = threadIdx.x & 31;
    int c = lane * 2;
    long long base = (long long)wave * CHANNEL + c;
    float dn = fmaxf(counts[wave], 1.0f);
    float2 v = *(float2*)(eagg + base);
    v.x /= dn; v.y /= dn;
    float sq = v.x * v.x + v.y * v.y;
    for (int m = 1; m < 32; m <<= 1) sq += __shfl_xor(sq, m, 32);
    float inv = 1.0f / fmaxf(sqrtf(sq), 1e-12f);
    v.x *= inv; v.y *= inv;
    *(float2*)(eagg + base) = v;
    float2 r0 = *(float2*)(eres + base);
    r0.x += v.x; r0.y += v.y;
    *(float2*)(eres + base) = r0;
}

// user: score = softmax(usrc @ latent^T) ; attn = score @ disen_w via V_WMMA_F32_16X16X4_F32 ;
//       v = uagg*(1+attn) ; l2-normalize rows ; uagg <- normalized (new emb) ; ures += normalized
// One wave32 handles 16 users (M=16). A tile = 16x4 score (2 VGPRs, ISA layout:
// lanes 0-15 hold K=0,1 ; lanes 16-31 hold K=2,3). B tile = 4x16 disen_w columns.
__global__ void k_user_finalize_wmma(const float* __restrict__ usrc,   // user emb for score
                                     const float* __restrict__ latent, // [4][64]
                                     const float* __restrict__ dw,     // disen_w [4][64]
                                     float* __restrict__ uagg,         // in: agg, out: new emb
                                     float* __restrict__ ures) {
    int waveId = blockIdx.x * (blockDim.x >> 5) + (threadIdx.x >> 5);
    int base = waveId * 16;
    if (base >= N_USERS) return;               // wave-uniform: EXEC stays all-ones for WMMA
    int lane = threadIdx.x & 31;
    int row  = lane & 15;                      // M index (A) / N index (B,D)
    int hi   = lane >> 4;                      // K half select
    int u    = base + row;                     // N_USERS % 16 == 0 -> always in range
    int k0   = 2 * hi, k1 = k0 + 1;

    // dots: score logits for factors k0,k1 of this row
    float d0 = 0.f, d1 = 0.f;
    const float* up = usrc + (long long)u * CHANNEL;
    for (int cc = 0; cc < CHANNEL; cc += 4) {
        float4 uv = *(const float4*)(up + cc);
        float4 l0 = *(const float4*)(latent + k0 * CHANNEL + cc);
        float4 l1 = *(const float4*)(latent + k1 * CHANNEL + cc);
        d0 += uv.x * l0.x + uv.y * l0.y + uv.z * l0.z + uv.w * l0.w;
        d1 += uv.x * l1.x + uv.y * l1.y + uv.z * l1.z + uv.w * l1.w;
    }
    // softmax over the 4 factors: partner half-wave holds the other two logits
    float o0 = __shfl_xor(d0, 16, 32);
    float o1 = __shfl_xor(d1, 16, 32);
    float m  = fmaxf(fmaxf(d0, d1), fmaxf(o0, o1));
    float e0 = __expf(d0 - m), e1 = __expf(d1 - m);
    float s  = e0 + e1 + __expf(o0 - m) + __expf(o1 - m);

    v2f A;
    A.x = e0 / s;   // score[row][k0]  -> A VGPR0 (K=0 lanes 0-15 / K=2 lanes 16-31)
    A.y = e1 / s;   // score[row][k1]  -> A VGPR1 (K=1 / K=3)

    // attn[16][64] = A(16x4) @ disen_w(4x64) as 4 N-tiles of 16
    v8f D[4];
    for (int t = 0; t < 4; ++t) {
        v2f B;
        B.x = dw[(2 * hi + 0) * CHANNEL + 16 * t + row];  // B row K=2*hi,   N=row
        B.y = dw[(2 * hi + 1) * CHANNEL + 16 * t + row];  // B row K=2*hi+1, N=row
        v8f C = {};
        D[t] = __builtin_amdgcn_wmma_f32_16x16x4_f32(false, A, false, B, (short)0, C,
                                                     false, false);
    }

    // D layout: element r of lane -> M = r + 8*hi, N = row (per 16x16 f32 C/D table)
    float vals[8][4];
    float sq[8];
    for (int r = 0; r < 8; ++r) {
        int rrow = base + r + 8 * hi;
        float p = 0.f;
        for (int t = 0; t < 4; ++t) {
            int col = 16 * t + row;
            float g = uagg[(long long)rrow * CHANNEL + col];
            float v = g * (1.0f + D[t][r]);     // user_agg*attn + user_agg
            vals[r][t] = v;
            p += v * v;
        }
        for (int mm = 1; mm < 16; mm <<= 1)     // row-wise sumsq across half-wave
            p += __shfl_xor(p, mm, 32);
        sq[r] = p;
    }
    for (int r = 0; r < 8; ++r) {
        int rrow = base + r + 8 * hi;
        float inv = 1.0f / fmaxf(sqrtf(sq[r]), 1e-12f);
        for (int t = 0; t < 4; ++t) {
            int col = 16 * t + row;
            float o = vals[r][t] * inv;
            long long idx = (long long)rrow * CHANNEL + col;
            uagg[idx] = o;                      // new user_emb for next hop
            ures[idx] += o;
        }
    }
}

// ---------------- host orchestration ----------------

extern "C" void kernel_launch(void* const* d_in, const int* in_sizes, int n_in,
                              void* d_out, int out_size, void* d_ws, size_t ws_size,
                              hipStream_t stream) {
    const float* user_in   = (const float*)d_in[0];   // [100000,64]
    const float* entity_in = (const float*)d_in[1];   // [200000,64]
    const float* latent    = (const float*)d_in[2];   // [4,64]
    const int*   head      = (const int*)d_in[3];     // edge_index[0]
    const int*   tail      = head + N_KG;             // edge_index[1]
    const int*   etype     = (const int*)d_in[4];     // [2M]
    const int*   uix       = (const int*)d_in[5];     // inter_edge[0]
    const int*   iix       = uix + N_INTER;           // inter_edge[1]
    const float* interw    = (const float*)d_in[6];   // [1M]
    const float* weight    = (const float*)d_in[7];   // [9,64]
    const float* att       = (const float*)d_in[8];   // [4,9]

    float* out_entity = (float*)d_out;                              // [200000*64]
    float* out_user   = out_entity + (long long)N_ENTITIES * CHANNEL; // [100000*64]
    float* out_cor    = out_user + (long long)N_USERS * CHANNEL;      // [1]

    // workspace carve-out (256B aligned)
    char* ws = (char*)d_ws;
    size_t off = 0;
    auto carve = [&](size_t bytes) -> float* {
        float* p = (float*)(ws + off);
        off = (off + bytes + 255) & ~(size_t)255;
        return p;
    };
    float* counts = carve((size_t)N_ENTITIES * 4);
    float* disenw = carve((size_t)N_FACTORS * CHANNEL * 4);
    float* E0     = carve((size_t)N_ENTITIES * CHANNEL * 4);
    float* E1     = carve((size_t)N_ENTITIES * CHANNEL * 4);
    float* U0     = carve((size_t)N_USERS * CHANNEL * 4);
    float* U1     = carve((size_t)N_USERS * CHANNEL * 4);

    const long long nE = (long long)N_ENTITIES * CHANNEL;
    const long long nU = (long long)N_USERS * CHANNEL;

    // init: zero hop-0 accumulators + counts, residuals = input embeddings
    k_zero_f32<<<4096, 256, 0, stream>>>(counts, N_ENTITIES);
    k_zero_f32<<<8192, 256, 0, stream>>>(E0, nE);
    k_zero_f32<<<8192, 256, 0, stream>>>(U0, nU);
    k_copy_f32<<<8192, 256, 0, stream>>>(out_entity, entity_in, nE);
    k_copy_f32<<<8192, 256, 0, stream>>>(out_user, user_in, nU);

    k_count_heads<<<(N_KG + 255) / 256, 256, 0, stream>>>(head, counts);
    k_disen_and_cor<<<1, 64, 0, stream>>>(att, weight, disenw, out_cor);

    const int kgBlocks    = (int)(((long long)N_KG * 32 + 255) / 256);
    const int interBlocks = (int)(((long long)N_INTER * 32 + 255) / 256);
    const int entBlocks   = (N_ENTITIES + 7) / 8;   // 8 waves/block, 1 entity/wave
    const int usrBlocks   = (N_USERS / 16 + 7) / 8; // 8 waves/block, 16 users/wave

    // ---- hop 0 (sources: input embeddings) ----
    k_kg_scatter<<<kgBlocks, 256, 0, stream>>>(head, tail, etype, entity_in, weight, E0);
    k_inter_scatter<<<interBlocks, 256, 0, stream>>>(uix, iix, interw, entity_in, U0);
    k_entity_finalize<<<entBlocks, 256, 0, stream>>>(E0, counts, out_entity);
    k_user_finalize_wmma<<<usrBlocks, 256, 0, stream>>>(user_in, latent, disenw, U0, out_user);

    // ---- hop 1 (sources: hop-0 normalized embeddings E0/U0) ----
    k_zero_f32<<<8192, 256, 0, stream>>>(E1, nE);
    k_zero_f32<<<8192, 256, 0, stream>>>(U1, nU);
    k_kg_scatter<<<kgBlocks, 256, 0, stream>>>(head, tail, etype, E0, weight, E1);
    k_inter_scatter<<<interBlocks, 256, 0, stream>>>(uix, iix, interw, E0, U1);
    k_entity_finalize<<<entBlocks, 256, 0, stream>>>(E1, counts, out_entity);
    k_user_finalize_wmma<<<usrBlocks, 256, 0, stream>>>(U0, latent, disenw, U1, out_user);
}